// Attn_40913858461777
// MI455X (gfx1250) — compile-verified
//
#include <hip/hip_runtime.h>
#include <math.h>

#define SEQ   32768
#define HID   1024
#define NPART 8          // d-dim partitions for v = W^T h

typedef float v2f __attribute__((ext_vector_type(2)));
typedef float v4f __attribute__((ext_vector_type(4)));
typedef float v8f __attribute__((ext_vector_type(8)));

// ---------------------------------------------------------------------------
// Phase 1a: vpart[p][j] = sum_{d in part p} W[d][j] * h[d]
// W is row-major (d, j); consecutive j across threads -> fully coalesced.
// ---------------------------------------------------------------------------
__global__ __launch_bounds__(256) void k_vpart(const float* __restrict__ W,
                                               const float* __restrict__ h,
                                               float* __restrict__ vpart) {
  const int j  = blockIdx.x * 256 + threadIdx.x;     // 0..1023
  const int d0 = blockIdx.y * (HID / NPART);         // 128 d's per partition
  float s = 0.f;
#pragma unroll 4
  for (int d = 0; d < HID / NPART; ++d)
    s = fmaf(W[(size_t)(d0 + d) * HID + j], h[d0 + d], s);
  vpart[blockIdx.y * HID + j] = s;
}

// Phase 1b: v[j] = sum_p vpart[p][j]
__global__ __launch_bounds__(256) void k_vred(const float* __restrict__ vpart,
                                              float* __restrict__ v) {
  const int j = blockIdx.x * 256 + threadIdx.x;
  float s = 0.f;
#pragma unroll
  for (int p = 0; p < NPART; ++p) s += vpart[p * HID + j];
  v[j] = s;
}

// ---------------------------------------------------------------------------
// Phase 2: scores[s] = enc[s] . v  via V_WMMA_F32_16X16X4_F32.
// 8 waves/block, 16 rows/wave, 128 rows/block, 256 blocks.
//
// K-permuted tiling so every lane loads a contiguous float4 (b128):
//   iteration i covers K in [8i, 8i+8):
//     lanes 0-15  load enc[row][8i   .. 8i+3]   (A-matrix lane half K0/K1)
//     lanes 16-31 load enc[row][8i+4 .. 8i+7]   (A-matrix lane half K2/K3)
//   WMMA step A consumes .xy -> K-slots {8i,8i+1 | 8i+4,8i+5}
//   WMMA step B consumes .zw -> K-slots {8i+2,8i+3 | 8i+6,8i+7}
// B operand (all 16 columns = v[k]) is built from the SAME offsets out of LDS,
// so each enc[r][k] is multiplied by exactly v[k]; the K permutation is just a
// reassociation of the f32 dot product.
// ---------------------------------------------------------------------------
__global__ __launch_bounds__(256) void k_scores(const float* __restrict__ enc,
                                                const float* __restrict__ v,
                                                float* __restrict__ scores) {
  __shared__ __align__(16) float vlds[HID];
  const int tid = threadIdx.x;
#pragma unroll
  for (int i = 0; i < HID / 256; ++i) vlds[tid + i * 256] = v[tid + i * 256];
  __syncthreads();

  const int wave  = tid >> 5;
  const int lane  = tid & 31;
  const int row0  = blockIdx.x * 128 + wave * 16;
  const int m     = lane & 15;              // row within the 16-row tile
  const int koff  = (lane >> 4) * 4;        // 0 for lanes 0-15, 4 for lanes 16-31

  const float* __restrict__ arow = enc + (size_t)(row0 + m) * HID + koff;
  const float* __restrict__ vptr = vlds + koff;

  v8f acc = {0.f, 0.f, 0.f, 0.f, 0.f, 0.f, 0.f, 0.f};

  for (int kk = 0; kk < HID; kk += 64) {
    __builtin_prefetch(arow + kk + 256, 0, 1);   // global_prefetch_b8, 1KB ahead
#pragma unroll
    for (int k0 = kk; k0 < kk + 64; k0 += 8) {
      v4f a = *(const v4f*)(arow + k0);          // 16B: enc[row][k0+koff .. +3]
      v4f b = *(const v4f*)(vptr + k0);          // 16B: v[k0+koff .. +3] (LDS bcast)
      v2f a0 = {a.x, a.y}, a1 = {a.z, a.w};
      v2f b0 = {b.x, b.y}, b1 = {b.z, b.w};
      acc = __builtin_amdgcn_wmma_f32_16x16x4_f32(
          false, a0, false, b0, (short)0, acc, false, false);
      acc = __builtin_amdgcn_wmma_f32_16x16x4_f32(
          false, a1, false, b1, (short)0, acc, false, false);
    }
  }

  // All N columns identical. VGPR j holds M=j (lanes 0-15) and M=j+8 (lanes 16-31).
  int outm = -1;
  if (lane < 8)                       outm = lane;            // rows 0..7
  else if (lane >= 16 && lane < 24)   outm = 8 + (lane - 16); // rows 8..15
  if (outm >= 0) {
    const int j = outm & 7;
    float val = acc[0];
#pragma unroll
    for (int t = 1; t < 8; ++t)
      if (j == t) val = acc[t];
    scores[row0 + outm] = val;
  }
}

// ---------------------------------------------------------------------------
// Phase 3: softmax over 32768 scores (deterministic, no float atomics)
// ---------------------------------------------------------------------------
__global__ __launch_bounds__(1024) void k_max(const float* __restrict__ scores,
                                              float* __restrict__ gmax) {
  __shared__ float red[1024];
  const int tid = threadIdx.x;
  float m = -3.4e38f;
  for (int i = tid; i < SEQ; i += 1024) m = fmaxf(m, scores[i]);
  red[tid] = m;
  __syncthreads();
  for (int s = 512; s > 0; s >>= 1) {
    if (tid < s) red[tid] = fmaxf(red[tid], red[tid + s]);
    __syncthreads();
  }
  if (tid == 0) gmax[0] = red[0];
}

__global__ __launch_bounds__(256) void k_exp(float* __restrict__ scores,
                                             const float* __restrict__ gmax,
                                             float* __restrict__ bsum) {
  __shared__ float red[256];
  const int tid = threadIdx.x;
  const int i   = blockIdx.x * 256 + tid;
  const float e = expf(scores[i] - gmax[0]);
  scores[i] = e;                       // in-place: now holds exp values
  red[tid] = e;
  __syncthreads();
  for (int s = 128; s > 0; s >>= 1) {
    if (tid < s) red[tid] += red[tid + s];
    __syncthreads();
  }
  if (tid == 0) bsum[blockIdx.x] = red[0];
}

__global__ __launch_bounds__(128) void k_sum(const float* __restrict__ bsum,
                                             float* __restrict__ gsum) {
  __shared__ float red[128];
  const int tid = threadIdx.x;
  red[tid] = bsum[tid];
  __syncthreads();
  for (int s = 64; s > 0; s >>= 1) {
    if (tid < s) red[tid] += red[tid + s];
    __syncthreads();
  }
  if (tid == 0) gsum[0] = red[0];
}

__global__ __launch_bounds__(256) void k_norm(const float* __restrict__ e,
                                              const float* __restrict__ gsum,
                                              float* __restrict__ out) {
  const int i = blockIdx.x * 256 + threadIdx.x;
  out[i] = e[i] / gsum[0];
}

// ---------------------------------------------------------------------------
// Launcher. Inputs: [0]=decoder_hidden(1024 f32), [1]=encoder_outputs
// (32768*1024 f32), [2]=W(1024*1024 f32), [3]=b(1024 f32, cancels in softmax).
// Output: 32768 f32.
// ---------------------------------------------------------------------------
extern "C" void kernel_launch(void* const* d_in, const int* in_sizes, int n_in,
                              void* d_out, int out_size, void* d_ws, size_t ws_size,
                              hipStream_t stream) {
  const float* h   = (const float*)d_in[0];
  const float* enc = (const float*)d_in[1];
  const float* W   = (const float*)d_in[2];
  float*       out = (float*)d_out;

  float* ws     = (float*)d_ws;
  float* vpart  = ws;                     // NPART*1024      = 8192
  float* v      = ws + 8192;              // 1024
  float* scores = ws + 9216;              // 32768
  float* bsum   = ws + 41984;             // 128
  float* gmax   = ws + 42112;             // 1
  float* gsum   = ws + 42113;             // 1   (~165 KB total)

  k_vpart <<<dim3(HID / 256, NPART), 256, 0, stream>>>(W, h, vpart);
  k_vred  <<<HID / 256,              256, 0, stream>>>(vpart, v);
  k_scores<<<SEQ / 128,              256, 0, stream>>>(enc, v, scores);
  k_max   <<<1,                     1024, 0, stream>>>(scores, gmax);
  k_exp   <<<SEQ / 256,              256, 0, stream>>>(scores, gmax, bsum);
  k_sum   <<<1,                      128, 0, stream>>>(bsum, gsum);
  k_norm  <<<SEQ / 256,              256, 0, stream>>>(scores, gsum, out);
}